// GroupedVectorQuantizer_21586505629901
// MI455X (gfx1250) — compile-verified
//
#include <hip/hip_runtime.h>

typedef __attribute__((ext_vector_type(2))) float v2f;
typedef __attribute__((ext_vector_type(8))) float v8f;

#define NGROUPS 8
#define NCODES  1024
#define GDIM    64
#define NROWS   32768
#define DDIM    (NGROUPS * GDIM) /* 512 */
#define NBLOCKS ((NROWS / 16) * NGROUPS) /* 16384 */

// ---------------------------------------------------------------------------
// Kernel 1: embed_sq[g,c] = sum_d embedding[g,c,d]^2   (8192 values, trivial)
// ---------------------------------------------------------------------------
__global__ void vq_embed_sq(const float* __restrict__ emb, float* __restrict__ esq) {
    int i = blockIdx.x * blockDim.x + threadIdx.x; // 0..8191
    if (i >= NGROUPS * NCODES) return;
    const float* e = emb + (size_t)i * GDIM;
    float s = 0.f;
#pragma unroll 16
    for (int d = 0; d < GDIM; ++d) s += e[d] * e[d];
    esq[i] = s;
}

// ---------------------------------------------------------------------------
// Kernel 2: main. One wave per (16-row tile, group).
//   scores = X_tile(16x64) @ E_g^T(64xC) via V_WMMA_F32_16X16X4_F32
//   argmin_c (e_sq[c] - 2*score), gather winning rows, partial MSE.
//   B tiles double-buffered in registers so WMMAs never wait per-load.
// ---------------------------------------------------------------------------
__global__ __launch_bounds__(32) void vq_main(const float* __restrict__ X,
                                              const float* __restrict__ E,
                                              const float* __restrict__ Esq,
                                              float* __restrict__ out,
                                              float* __restrict__ partial) {
    const int g      = blockIdx.y;        // group 0..7
    const int row0   = blockIdx.x * 16;   // row tile base
    const int lane   = threadIdx.x;       // 0..31
    const int lane_c = lane & 15;         // row/code within half-wave
    const int lane_k = (lane >> 4) * 2;   // K offset {0,2} per half-wave

    const float* Eg   = E   + (size_t)g * NCODES * GDIM;
    const float* EsqG = Esq + g * NCODES;

    // ---- A tile (16 rows x 64 K) resident in registers: 16 x float2 -------
    const float* Arow = X + (size_t)(row0 + lane_c) * DDIM + g * GDIM + lane_k;
    v2f a[16];
#pragma unroll
    for (int k = 0; k < 16; ++k) a[k] = *(const v2f*)(Arow + 4 * k);

    float minv[8];
    int   mini[8];
#pragma unroll
    for (int j = 0; j < 8; ++j) { minv[j] = 3.0e38f; mini[j] = 0; }

    const float* Bbase = Eg + (size_t)lane_c * GDIM + lane_k;

    // 16 WMMAs (two independent chains) + argmin update for one column tile
    auto compute_tile = [&](const v2f (&b)[16], float esq, int c0) {
        v8f acc0 = {0.f, 0.f, 0.f, 0.f, 0.f, 0.f, 0.f, 0.f};
        v8f acc1 = {0.f, 0.f, 0.f, 0.f, 0.f, 0.f, 0.f, 0.f};
#pragma unroll
        for (int k = 0; k < 8; ++k) {
            acc0 = __builtin_amdgcn_wmma_f32_16x16x4_f32(
                false, a[k], false, b[k], (short)0, acc0, false, false);
            acc1 = __builtin_amdgcn_wmma_f32_16x16x4_f32(
                false, a[k + 8], false, b[k + 8], (short)0, acc1, false, false);
        }
        const int idx = c0 + lane_c;
#pragma unroll
        for (int j = 0; j < 8; ++j) {
            float v   = esq - 2.0f * (acc0[j] + acc1[j]);
            bool  hit = v < minv[j];
            minv[j] = hit ? v   : minv[j];
            mini[j] = hit ? idx : mini[j];
        }
    };

    auto load_tile = [&](v2f (&b)[16], float& esq, int t) {
        const float* Bp = Bbase + (size_t)t * 16 * GDIM;
#pragma unroll
        for (int k = 0; k < 16; ++k) b[k] = *(const v2f*)(Bp + 4 * k);
        esq = EsqG[t * 16 + lane_c];
    };

    // ---- ping-pong over 64 column tiles (even count, no copies) -----------
    v2f b0[16], b1[16];
    float esq0, esq1;
    load_tile(b0, esq0, 0);
    for (int t = 0; t < 62; t += 2) {
        load_tile(b1, esq1, t + 1);      // prefetch t+1 while computing t
        compute_tile(b0, esq0, t * 16);
        load_tile(b0, esq0, t + 2);      // prefetch t+2 while computing t+1
        compute_tile(b1, esq1, (t + 1) * 16);
    }
    load_tile(b1, esq1, 63);
    compute_tile(b0, esq0, 62 * 16);
    compute_tile(b1, esq1, 63 * 16);

    // ---- cross-lane argmin over 16 columns (xor stays within half-wave) ---
#pragma unroll
    for (int j = 0; j < 8; ++j) {
#pragma unroll
        for (int s = 1; s < 16; s <<= 1) {
            float ov = __shfl_xor(minv[j], s, 32);
            int   oi = __shfl_xor(mini[j], s, 32);
            bool take = (ov < minv[j]) || (ov == minv[j] && oi < mini[j]);
            minv[j] = take ? ov : minv[j];
            mini[j] = take ? oi : mini[j];
        }
    }
    // lanes 0-15 now hold argmin for rows 0..7 (VGPR j), lanes 16-31 rows 8..15

    // ---- gather winning code rows, write outputs, accumulate MSE ----------
    float* qout = out;
    float* iout = out + (size_t)NROWS * DDIM;
    float  lsum = 0.f;
#pragma unroll
    for (int r = 0; r < 16; ++r) {
        int idx = (r < 8) ? __shfl(mini[r], 0, 32) : __shfl(mini[r - 8], 16, 32);
        const float* qrow = Eg + (size_t)idx * GDIM + 2 * lane;
        const float* xrow = X + (size_t)(row0 + r) * DDIM + g * GDIM + 2 * lane;
        v2f q = *(const v2f*)qrow;
        v2f x = *(const v2f*)xrow;
        *(v2f*)(qout + (size_t)(row0 + r) * DDIM + g * GDIM + 2 * lane) = q;
        float dx = q.x - x.x, dy = q.y - x.y;
        lsum += dx * dx + dy * dy;
        if (lane == 0) iout[(size_t)(row0 + r) * NGROUPS + g] = (float)idx;
    }
#pragma unroll
    for (int s = 1; s < 32; s <<= 1) lsum += __shfl_xor(lsum, s, 32);
    if (lane == 0) partial[blockIdx.y * gridDim.x + blockIdx.x] = lsum;
}

// ---------------------------------------------------------------------------
// Kernel 3: deterministic reduction of 16384 partials -> 3 scalar losses
// ---------------------------------------------------------------------------
__global__ void vq_finalize(const float* __restrict__ partial, float* __restrict__ out) {
    __shared__ float sm[256];
    int t = threadIdx.x;
    float s = 0.f;
    for (int i = t; i < NBLOCKS; i += 256) s += partial[i]; // fixed order
    sm[t] = s;
    __syncthreads();
    for (int w = 128; w > 0; w >>= 1) {
        if (t < w) sm[t] += sm[t + w];
        __syncthreads();
    }
    if (t == 0) {
        float mse = sm[0] / (float)((size_t)NROWS * DDIM);
        float* lp = out + (size_t)NROWS * DDIM + (size_t)NROWS * NGROUPS;
        lp[0] = mse * 1.25f; // vq_loss = codebook + 0.25*commit
        lp[1] = mse;         // codebook_loss
        lp[2] = mse;         // commit_loss
    }
}

// ---------------------------------------------------------------------------
extern "C" void kernel_launch(void* const* d_in, const int* in_sizes, int n_in,
                              void* d_out, int out_size, void* d_ws, size_t ws_size,
                              hipStream_t stream) {
    (void)in_sizes; (void)n_in; (void)out_size; (void)ws_size;
    const float* X = (const float*)d_in[0]; // inputs   (32768 x 512)
    const float* E = (const float*)d_in[1]; // embedding (8 x 1024 x 64)
    float* out = (float*)d_out;
    float* ws  = (float*)d_ws;
    float* partial = ws;           // 16384 floats
    float* esq     = ws + NBLOCKS; // 8192 floats

    vq_embed_sq<<<(NGROUPS * NCODES + 255) / 256, 256, 0, stream>>>(E, esq);

    dim3 grid(NROWS / 16, NGROUPS);
    vq_main<<<grid, 32, 0, stream>>>(X, E, esq, out, partial);

    vq_finalize<<<1, 256, 0, stream>>>(partial, out);
}